// GlobalGATOverClusters_3925600109369
// MI455X (gfx1250) — compile-verified
//
#include <hip/hip_runtime.h>
#include <hip/hip_bf16.h>

#define IN_DIM 128
#define OUT_DIM 32
#define NHEAD 4
#define EDIM 16
#define HID 128   // NHEAD*OUT_DIM
#define NEG_SLOPE 0.2f

typedef __attribute__((ext_vector_type(2))) float v2f;
typedef __attribute__((ext_vector_type(8))) float v8f;

// exact parameter types for the async-to-LDS builtin (from hipcc diagnostic):
// pointer to GCC-vector int4, in AS1 (global) / AS3 (LDS)
typedef int gvi4 __attribute__((vector_size(16)));
typedef __attribute__((address_space(1))) gvi4* as1_vi4p;
typedef __attribute__((address_space(3))) gvi4* as3_vi4p;

#if __has_builtin(__builtin_amdgcn_global_load_async_to_lds_b128)
#define HAVE_ASYNC_LDS 1
#else
#define HAVE_ASYNC_LDS 0
#endif

__device__ __forceinline__ void wait_async0() {
#if __has_builtin(__builtin_amdgcn_s_wait_asynccnt)
    __builtin_amdgcn_s_wait_asynccnt(0);
#else
    asm volatile("s_wait_asynccnt 0x0" ::: "memory");
#endif
}

// ---- order-preserving float <-> uint encoding for atomicMax ----
__device__ __forceinline__ unsigned encf(float f) {
    unsigned u = __float_as_uint(f);
    return (u & 0x80000000u) ? ~u : (u | 0x80000000u);
}
__device__ __forceinline__ float decf(unsigned u) {
    unsigned b = (u & 0x80000000u) ? (u & 0x7FFFFFFFu) : ~u;
    return __uint_as_float(b);
}

// ---- fold att_edge into W_edge: M[d][h] = sum_c W_edge[d, h*32+c] * att_edge[h,c] ----
__global__ void k_medge(const float* __restrict__ W_edge,
                        const float* __restrict__ att_edge,
                        float* __restrict__ M) {
    int t = threadIdx.x;
    if (t >= EDIM * NHEAD) return;
    int d = t >> 2, h = t & 3;
    float s = 0.f;
    for (int c = 0; c < OUT_DIM; ++c)
        s += W_edge[d * HID + h * OUT_DIM + c] * att_edge[h * OUT_DIM + c];
    M[d * NHEAD + h] = s;
}

// ---- xh = x @ W via v_wmma_f32_16x16x4_f32 ----
// One block = 256 threads = 8 waves. Block owns a 16-row strip of x (staged in
// LDS, asynchronously on gfx1250); wave w computes the 16x16 output tile at
// columns [w*16, w*16+16).
__global__ __launch_bounds__(256) void k_gemm(const float* __restrict__ x,
                                              const float* __restrict__ W,
                                              float* __restrict__ xh, int n) {
    __shared__ float sx[16 * IN_DIM];
    const int tid = threadIdx.x;
    const int row0 = blockIdx.x * 16;

#if HAVE_ASYNC_LDS
    // async copy of the 16x128 f32 tile: 512 x b128 chunks, 2 per thread,
    // memory -> LDS without touching VGPRs (ASYNCcnt-tracked).
    for (int ci = tid; ci < 512; ci += 256) {
        int r = ci >> 5, c4 = ci & 31;       // row, 4-float chunk within row
        int gr = row0 + r;
        if (gr >= n) gr = n - 1;             // clamp OOB rows (stores guarded)
        const float* gp = x + (size_t)gr * IN_DIM + c4 * 4;
        __builtin_amdgcn_global_load_async_to_lds_b128(
            (as1_vi4p)gp, (as3_vi4p)&sx[ci * 4], 0, 0);
    }
    wait_async0();
    __syncthreads();
#else
    for (int i = tid; i < 16 * IN_DIM; i += 256) {
        int r = i >> 7, c = i & 127;
        int gr = row0 + r;
        if (gr >= n) gr = n - 1;
        sx[i] = x[(size_t)gr * IN_DIM + c];
    }
    __syncthreads();
#endif

    const int wave = tid >> 5;
    const int lane = tid & 31;
    const int half = lane >> 4;   // 0: lanes 0-15, 1: lanes 16-31
    const int l    = lane & 15;
    const int col0 = wave * 16;

    v8f acc = {};
    for (int k = 0; k < IN_DIM; k += 4) {
        // A 16x4 (M=row l, VGPR0 = K {k,k+2}[half], VGPR1 = +1)  [ISA 7.12.2]
        v2f a, b;
        a.x = sx[l * IN_DIM + k + 2 * half];
        a.y = sx[l * IN_DIM + k + 2 * half + 1];
        // B 4x16 (N=col0+l), same half-split K layout
        b.x = W[(size_t)(k + 2 * half) * HID + col0 + l];
        b.y = W[(size_t)(k + 2 * half + 1) * HID + col0 + l];
        acc = __builtin_amdgcn_wmma_f32_16x16x4_f32(
            false, a, false, b, (short)0, acc, false, false);
    }
    // C/D: VGPR r -> M = r + 8*half, N = l   [ISA 7.12.2]
    for (int r = 0; r < 8; ++r) {
        int m = row0 + r + 8 * half;
        if (m < n) xh[(size_t)m * HID + col0 + l] = acc[r];
    }
}

// ---- a_src[n,h], a_dst[n,h] from xh ----
__global__ void k_nodeatt(const float* __restrict__ xh,
                          const float* __restrict__ att_src,
                          const float* __restrict__ att_dst,
                          float* __restrict__ a_src, float* __restrict__ a_dst,
                          int n) {
    int i = blockIdx.x * blockDim.x + threadIdx.x;
    if (i >= n * NHEAD) return;
    int nn = i >> 2, h = i & 3;
    const float* p = xh + (size_t)nn * HID + h * OUT_DIM;
    const float* s1 = att_src + h * OUT_DIM;
    const float* s2 = att_dst + h * OUT_DIM;
    float as = 0.f, ad = 0.f;
    for (int c = 0; c < OUT_DIM; ++c) { float v = p[c]; as += v * s1[c]; ad += v * s2[c]; }
    a_src[i] = as;
    a_dst[i] = ad;
}

// ---- self-loop attr: sum incoming edge_attr + count per dst ----
__global__ void k_loopsum(const int* __restrict__ dst,
                          const float* __restrict__ ea,
                          float* __restrict__ cnt, float* __restrict__ loop,
                          int e) {
    int i = blockIdx.x * blockDim.x + threadIdx.x;
    if (i >= e) return;
    int d = dst[i];
    atomicAdd(cnt + d, 1.0f);
    const float* a = ea + (size_t)i * EDIM;
    float* lp = loop + (size_t)d * EDIM;
    for (int j = 0; j < EDIM; ++j) atomicAdd(lp + j, a[j]);
}

__global__ void k_loopdiv(float* __restrict__ loop, const float* __restrict__ cnt, int n) {
    int i = blockIdx.x * blockDim.x + threadIdx.x;
    if (i >= n * EDIM) return;
    float c = cnt[i >> 4];
    loop[i] /= fmaxf(c, 1.0f);
}

// ---- pass 1: raw attention logits + leaky relu + segment max ----
__global__ void k_alpha1(const int* __restrict__ src, const int* __restrict__ dst,
                         const float* __restrict__ ea, const float* __restrict__ loopattr,
                         const float* __restrict__ M,
                         const float* __restrict__ a_src, const float* __restrict__ a_dst,
                         float* __restrict__ alpha, unsigned* __restrict__ maxenc,
                         int e, int n) {
    int i = blockIdx.x * blockDim.x + threadIdx.x;
    if (i >= e + n) return;
    int s, d;
    const float* ap;
    if (i < e) { s = src[i]; d = dst[i]; ap = ea + (size_t)i * EDIM; }
    else       { s = d = i - e;          ap = loopattr + (size_t)s * EDIM; }
    float acc[NHEAD] = {0.f, 0.f, 0.f, 0.f};
    for (int j = 0; j < EDIM; ++j) {
        float v = ap[j];
        acc[0] += v * M[j * NHEAD + 0];
        acc[1] += v * M[j * NHEAD + 1];
        acc[2] += v * M[j * NHEAD + 2];
        acc[3] += v * M[j * NHEAD + 3];
    }
    for (int h = 0; h < NHEAD; ++h) {
        float al = a_src[s * NHEAD + h] + a_dst[d * NHEAD + h] + acc[h];
        al = (al > 0.f) ? al : NEG_SLOPE * al;
        alpha[(size_t)i * NHEAD + h] = al;
        atomicMax(maxenc + (size_t)d * NHEAD + h, encf(al));
    }
}

// ---- pass 2: exp(alpha - max) and segment denominator ----
__global__ void k_alpha2(const int* __restrict__ dst,
                         float* __restrict__ alpha,
                         const unsigned* __restrict__ maxenc,
                         float* __restrict__ den, int e, int n) {
    int i = blockIdx.x * blockDim.x + threadIdx.x;
    if (i >= e + n) return;
    int d = (i < e) ? dst[i] : (i - e);
    for (int h = 0; h < NHEAD; ++h) {
        float al = alpha[(size_t)i * NHEAD + h];
        float mx = decf(maxenc[(size_t)d * NHEAD + h]);
        float ex = __expf(al - mx);
        alpha[(size_t)i * NHEAD + h] = ex;
        atomicAdd(den + (size_t)d * NHEAD + h, ex);
    }
}

// ---- pass 2.5: normalize alpha in place (4 divides/edge instead of 128) ----
__global__ void k_norm(const int* __restrict__ dst,
                       float* __restrict__ alpha,
                       const float* __restrict__ den, int e, int n) {
    int i = blockIdx.x * blockDim.x + threadIdx.x;
    if (i >= e + n) return;
    int d = (i < e) ? dst[i] : (i - e);
    for (int h = 0; h < NHEAD; ++h)
        alpha[(size_t)i * NHEAD + h] /= den[(size_t)d * NHEAD + h];
}

// ---- pass 3: weighted scatter-aggregate. One wave32 per edge; lane = column
// within each head. Prefetch gathered xh row early (global_prefetch_b8). ----
__global__ __launch_bounds__(256) void k_scatter(const int* __restrict__ src,
                                                 const int* __restrict__ dst,
                                                 const float* __restrict__ xh,
                                                 const float* __restrict__ alpha,
                                                 float* __restrict__ acc,
                                                 int e, int n) {
    int gw = (int)((blockIdx.x * blockDim.x + threadIdx.x) >> 5);
    int lane = threadIdx.x & 31;
    int total = e + n;
    if (gw >= total) return;
    int s, d;
    if (gw < e) { s = src[gw]; d = dst[gw]; } else { s = d = gw - e; }
    const float* xr = xh + (size_t)s * HID;
    __builtin_prefetch(xr, 0, 0);           // -> global_prefetch_b8
    float* ar = acc + (size_t)d * HID;
    for (int h = 0; h < NHEAD; ++h) {
        float w = alpha[(size_t)gw * NHEAD + h];
        atomicAdd(ar + h * OUT_DIM + lane, xr[h * OUT_DIM + lane] * w);
    }
}

// ---- head mean + bias ----
__global__ void k_final(const float* __restrict__ acc, const float* __restrict__ bias,
                        float* __restrict__ out, int n) {
    int i = blockIdx.x * blockDim.x + threadIdx.x;
    if (i >= n * OUT_DIM) return;
    int nn = i >> 5, c = i & 31;
    const float* a = acc + (size_t)nn * HID + c;
    out[i] = 0.25f * (a[0] + a[OUT_DIM] + a[2 * OUT_DIM] + a[3 * OUT_DIM]) + bias[c];
}

extern "C" void kernel_launch(void* const* d_in, const int* in_sizes, int n_in,
                              void* d_out, int out_size, void* d_ws, size_t ws_size,
                              hipStream_t stream) {
    const float* x        = (const float*)d_in[0];
    const int*   ei       = (const int*)  d_in[1];
    const float* ea       = (const float*)d_in[2];
    const float* W        = (const float*)d_in[3];
    const float* att_src  = (const float*)d_in[4];
    const float* att_dst  = (const float*)d_in[5];
    const float* W_edge   = (const float*)d_in[6];
    const float* att_edge = (const float*)d_in[7];
    const float* bias     = (const float*)d_in[8];
    float* out = (float*)d_out;

    const int n = in_sizes[0] / IN_DIM;
    const int e = in_sizes[2] / EDIM;
    const int* srcI = ei;
    const int* dstI = ei + e;

    // workspace layout (float units)
    float* ws = (float*)d_ws;
    size_t o_xh    = 0;
    size_t o_asrc  = o_xh    + (size_t)n * HID;
    size_t o_adst  = o_asrc  + (size_t)n * NHEAD;
    size_t o_alpha = o_adst  + (size_t)n * NHEAD;
    size_t o_M     = o_alpha + (size_t)(e + n) * NHEAD;
    size_t o_zero  = o_M     + 64;             // start of zero-init region
    size_t o_cnt   = o_zero;
    size_t o_loop  = o_cnt   + (size_t)n;
    size_t o_max   = o_loop  + (size_t)n * EDIM;
    size_t o_den   = o_max   + (size_t)n * NHEAD;
    size_t o_acc   = o_den   + (size_t)n * NHEAD;
    size_t o_end   = o_acc   + (size_t)n * HID;
    if (ws_size < o_end * sizeof(float)) return;   // scratch too small: no-op

    float*    w_xh    = ws + o_xh;
    float*    w_asrc  = ws + o_asrc;
    float*    w_adst  = ws + o_adst;
    float*    w_alpha = ws + o_alpha;
    float*    w_M     = ws + o_M;
    float*    w_cnt   = ws + o_cnt;
    float*    w_loop  = ws + o_loop;
    unsigned* w_max   = (unsigned*)(ws + o_max);
    float*    w_den   = ws + o_den;
    float*    w_acc   = ws + o_acc;

    // zero accumulators (enc-max init of 0 sorts below every real float)
    (void)hipMemsetAsync(ws + o_zero, 0, (o_end - o_zero) * sizeof(float), stream);

    k_medge<<<1, 64, 0, stream>>>(W_edge, att_edge, w_M);

    k_gemm<<<(n + 15) / 16, 256, 0, stream>>>(x, W, w_xh, n);

    k_nodeatt<<<(n * NHEAD + 255) / 256, 256, 0, stream>>>(w_xh, att_src, att_dst,
                                                           w_asrc, w_adst, n);

    k_loopsum<<<(e + 255) / 256, 256, 0, stream>>>(dstI, ea, w_cnt, w_loop, e);
    k_loopdiv<<<(n * EDIM + 255) / 256, 256, 0, stream>>>(w_loop, w_cnt, n);

    int tot = e + n;
    k_alpha1<<<(tot + 255) / 256, 256, 0, stream>>>(srcI, dstI, ea, w_loop, w_M,
                                                    w_asrc, w_adst, w_alpha, w_max, e, n);
    k_alpha2<<<(tot + 255) / 256, 256, 0, stream>>>(dstI, w_alpha, w_max, w_den, e, n);
    k_norm<<<(tot + 255) / 256, 256, 0, stream>>>(dstI, w_alpha, w_den, e, n);

    long long thr = (long long)tot * 32;
    k_scatter<<<(unsigned)((thr + 255) / 256), 256, 0, stream>>>(srcI, dstI, w_xh,
                                                                 w_alpha, w_acc, e, n);

    k_final<<<(n * OUT_DIM + 255) / 256, 256, 0, stream>>>(w_acc, bias, out, n);
}